// DWT2DInverse_9801115369937
// MI455X (gfx1250) — compile-verified
//
#include <hip/hip_runtime.h>
#include <cstdint>

// Inverse 2D DWT with 2-tap synthesis filters (Haar-class), mode='zero'.
//
// Derivation from the reference (conv_transpose, stride 2, pad L-2, L=2):
//   column stage:  lo[2r+i, c] = g0c[i]*LL[r,c] + g1c[i]*LH[r,c]
//                  hi[2r+i, c] = g0c[i]*HL[r,c] + g1c[i]*HH[r,c]
//   row stage:     y[R, 2c+j]  = g0r[j]*lo[R,c] + g1r[j]*hi[R,c]
// => each input (r,c) across the 4 subbands produces one 2x2 output block.
//
// Shapes: low (4,64,256,256) f32, highs (4,64,3,256,256) f32,
//         out (4,64,512,512) f32.  Pure streaming: 512 MB total traffic,
//         ~22 us roofline at 23.3 TB/s. No GEMM structure -> no WMMA; the
//         CDNA5 feature that applies is the async global->LDS mover + NT
//         cache policy (stream-once data vs 192 MB L2).

typedef float v4f __attribute__((ext_vector_type(4)));

#define NPLANES 256u          // N*C = 4*64
#define H_IN    256u
#define W_IN    256u
#define W4      (W_IN / 4u)   // quads per row

__global__ __launch_bounds__(256) void idwt2_tap2_kernel(
    const float* __restrict__ low,
    const float* __restrict__ highs,
    const float* __restrict__ g0c, const float* __restrict__ g1c,
    const float* __restrict__ g0r, const float* __restrict__ g1r,
    float* __restrict__ out, uint32_t nquads)
{
    // Per-subband staging: smem[subband*256 + tid], one float4 per thread.
    __shared__ v4f smem[4 * 256];

    const uint32_t tid = threadIdx.x;
    const uint32_t q   = blockIdx.x * 256u + tid;
    if (q >= nquads) return;

    const uint32_t w4 = q & (W4 - 1u);        // quad column
    const uint32_t t  = q / W4;
    const uint32_t h  = t & (H_IN - 1u);      // input row
    const uint32_t pl = t / H_IN;             // plane = n*64 + c

    const uint32_t pix    = h * W_IN + w4 * 4u;                      // elem in plane
    const uint32_t lowOff = (pl * (H_IN * W_IN) + pix) * 4u;         // bytes
    const uint32_t lhOff  = ((pl * 3u + 0u) * (H_IN * W_IN) + pix) * 4u;
    const uint32_t hlOff  = ((pl * 3u + 1u) * (H_IN * W_IN) + pix) * 4u;
    const uint32_t hhOff  = ((pl * 3u + 2u) * (H_IN * W_IN) + pix) * 4u;

    // Low 32 bits of the generic pointer to a __shared__ object are the
    // wave-relative LDS byte offset (ISA 10.2 aperture mapping) — exactly
    // what GLOBAL_LOAD_ASYNC_TO_LDS expects in its LDS-address VGPR.
    const uint32_t ldsBase = (uint32_t)(uintptr_t)(void*)smem;
    const uint32_t l0 = ldsBase + (0u * 256u + tid) * 16u;
    const uint32_t l1 = ldsBase + (1u * 256u + tid) * 16u;
    const uint32_t l2 = ldsBase + (2u * 256u + tid) * 16u;
    const uint32_t l3 = ldsBase + (3u * 256u + tid) * 16u;

    // Async 128-bit global->LDS copies (GVS mode: SGPR base + 32-bit VGPR
    // byte offset). Non-temporal: inputs are read exactly once.
    asm volatile("global_load_async_to_lds_b128 %0, %1, %2 th:TH_LOAD_NT"
                 :: "v"(l0), "v"(lowOff), "s"(low) : "memory");
    asm volatile("global_load_async_to_lds_b128 %0, %1, %2 th:TH_LOAD_NT"
                 :: "v"(l1), "v"(lhOff), "s"(highs) : "memory");
    asm volatile("global_load_async_to_lds_b128 %0, %1, %2 th:TH_LOAD_NT"
                 :: "v"(l2), "v"(hlOff), "s"(highs) : "memory");
    asm volatile("global_load_async_to_lds_b128 %0, %1, %2 th:TH_LOAD_NT"
                 :: "v"(l3), "v"(hhOff), "s"(highs) : "memory");

    // Uniform scalar filter taps (overlap with async copies in flight).
    const float a0 = g0c[0], a1 = g0c[1];   // column synthesis, low-pass
    const float b0 = g1c[0], b1 = g1c[1];   // column synthesis, high-pass
    const float c0 = g0r[0], c1 = g0r[1];   // row synthesis, low-pass
    const float d0 = g1r[0], d1 = g1r[1];   // row synthesis, high-pass

    // Each wave reads only its own tid-disjoint LDS slots -> no barrier,
    // just drain this wave's async copies.
    asm volatile("s_wait_asynccnt 0x0" ::: "memory");

    const v4f ll = smem[0 * 256 + tid];
    const v4f lh = smem[1 * 256 + tid];
    const v4f hl = smem[2 * 256 + tid];
    const v4f hh = smem[3 * 256 + tid];

    // Column stage (vertical), 4 pixels at a time.
    const v4f loE = a0 * ll + b0 * lh;   // output row 2h
    const v4f loO = a1 * ll + b1 * lh;   // output row 2h+1
    const v4f hiE = a0 * hl + b0 * hh;
    const v4f hiO = a1 * hl + b1 * hh;

    // Row stage (horizontal): even/odd output columns.
    const v4f ee = c0 * loE + d0 * hiE;  // (even row, even col)
    const v4f eo = c1 * loE + d1 * hiE;  // (even row, odd  col)
    const v4f oe = c0 * loO + d0 * hiO;  // (odd  row, even col)
    const v4f oo = c1 * loO + d1 * hiO;  // (odd  row, odd  col)

    // Interleave to 2 rows x 8 consecutive floats (two b128 stores per row).
    v4f r0a, r0b, r1a, r1b;
    r0a.x = ee.x; r0a.y = eo.x; r0a.z = ee.y; r0a.w = eo.y;
    r0b.x = ee.z; r0b.y = eo.z; r0b.z = ee.w; r0b.w = eo.w;
    r1a.x = oe.x; r1a.y = oo.x; r1a.z = oe.y; r1a.w = oo.y;
    r1b.x = oe.z; r1b.y = oo.z; r1b.z = oe.w; r1b.w = oo.w;

    const uint32_t Wo = 2u * W_IN;       // 512
    float* o0 = out + (size_t)pl * (size_t)(4u * H_IN * W_IN)
                    + (size_t)(2u * h) * Wo + (size_t)(w4 * 8u);

    // Output is written once and never re-read: stream past the caches.
    __builtin_nontemporal_store(r0a, (v4f*)o0);
    __builtin_nontemporal_store(r0b, (v4f*)(o0 + 4));
    __builtin_nontemporal_store(r1a, (v4f*)(o0 + Wo));
    __builtin_nontemporal_store(r1b, (v4f*)(o0 + Wo + 4));
}

extern "C" void kernel_launch(void* const* d_in, const int* in_sizes, int n_in,
                              void* d_out, int out_size, void* d_ws, size_t ws_size,
                              hipStream_t stream) {
    const float* low   = (const float*)d_in[0];
    const float* highs = (const float*)d_in[1];
    const float* g0c   = (const float*)d_in[2];
    const float* g1c   = (const float*)d_in[3];
    const float* g0r   = (const float*)d_in[4];
    const float* g1r   = (const float*)d_in[5];
    float* out = (float*)d_out;

    const uint32_t nquads = (uint32_t)(in_sizes[0] / 4);     // 4,194,304
    const uint32_t blocks = (nquads + 255u) / 256u;          // 16,384

    hipLaunchKernelGGL(idwt2_tap2_kernel, dim3(blocks), dim3(256), 0, stream,
                       low, highs, g0c, g1c, g0r, g1r, out, nquads);
}